// ComplexMixture_11536282157523
// MI455X (gfx1250) — compile-verified
//
#include <hip/hip_runtime.h>

typedef __attribute__((ext_vector_type(2))) float v2f;
typedef __attribute__((ext_vector_type(8))) float v8f;

#define S_LEN 1024
#define D_LEN 256
#define KB    32
#define LDK   34   // padded K stride (even -> 8B-aligned b64 LDS loads, conflict-free)

__global__ __launch_bounds__(256, 3) void complex_mixture_wmma(
    const float* __restrict__ in_r,
    const float* __restrict__ in_i,
    const float* __restrict__ wgt,
    float* __restrict__ out_r,
    float* __restrict__ out_i)
{
    __shared__ __align__(16) float sWrA[64  * LDK];  // (w*real)  A-tile, [m][k]
    __shared__ __align__(16) float sWiA[64  * LDK];  // (w*imag)  A-tile, [m][k]
    __shared__ __align__(16) float sRB [128 * LDK];  // real      B-tile, [n][k]
    __shared__ __align__(16) float sIB [128 * LDK];  // imag      B-tile, [n][k]

    const int b     = blockIdx.z;
    const int aBase = blockIdx.y * 64;    // output row range (a)
    const int cBase = blockIdx.x * 128;   // output col range (c)

    const int tid   = threadIdx.x;
    const int lane  = tid & 31;           // wave32
    const int wave  = tid >> 5;
    const int waveM = wave & 1;           // 2 x 32 rows
    const int waveN = wave >> 1;          // 4 x 32 cols

    const int laneLo = lane & 15;
    const int kOff   = (lane >> 4) * 2;   // lanes 0-15 -> k0,k0+1 ; lanes 16-31 -> k0+2,k0+3

    const float* rSrc = in_r + (size_t)b * S_LEN * D_LEN;
    const float* iSrc = in_i + (size_t)b * S_LEN * D_LEN;
    const float* wSrc = wgt  + (size_t)b * S_LEN;

    v8f accR [2][2] = {};   // out_r = (wr)^T r + (wi)^T i   (both chained into C)
    v8f accI1[2][2] = {};   // (wi)^T r
    v8f accI2[2][2] = {};   // (wr)^T i

    for (int kb = 0; kb < S_LEN / KB; ++kb) {
        const int s0 = kb * KB;

        // ---- stage weighted A tiles: 32(k) x 64(m), transposed to [m][k] ----
        #pragma unroll
        for (int it = 0; it < (KB * 64) / 256; ++it) {      // 8 iters
            int idx = tid + it * 256;
            int sl  = idx >> 6;                             // 0..31
            int ml  = idx & 63;
            size_t g = (size_t)(s0 + sl) * D_LEN + aBase + ml;
            float  w = wSrc[s0 + sl];
            sWrA[ml * LDK + sl] = w * rSrc[g];
            sWiA[ml * LDK + sl] = w * iSrc[g];
        }
        // ---- stage unweighted B tiles: 32(k) x 128(n), transposed to [n][k] ----
        #pragma unroll
        for (int it = 0; it < (KB * 128) / 256; ++it) {     // 16 iters
            int idx = tid + it * 256;
            int sl  = idx >> 7;                             // 0..31
            int nl  = idx & 127;
            size_t g = (size_t)(s0 + sl) * D_LEN + cBase + nl;
            sRB[nl * LDK + sl] = rSrc[g];
            sIB[nl * LDK + sl] = iSrc[g];
        }
        // prefetch next K-block of the A region (emits global_prefetch_b8)
        if (kb + 1 < S_LEN / KB) {
            size_t gn = (size_t)(s0 + KB + (tid >> 3)) * D_LEN + aBase + (tid & 7) * 8;
            __builtin_prefetch(&rSrc[gn], 0, 0);
            __builtin_prefetch(&iSrc[gn], 0, 0);
        }
        __syncthreads();

        #pragma unroll
        for (int kk = 0; kk < KB; kk += 4) {
            v2f aR[2], aI[2], bR[2], bI[2];
            #pragma unroll
            for (int mt = 0; mt < 2; ++mt) {
                int row = waveM * 32 + mt * 16 + laneLo;
                aR[mt] = *(const v2f*)&sWrA[row * LDK + kk + kOff];
                aI[mt] = *(const v2f*)&sWiA[row * LDK + kk + kOff];
            }
            #pragma unroll
            for (int nt = 0; nt < 2; ++nt) {
                int col = waveN * 32 + nt * 16 + laneLo;
                bR[nt] = *(const v2f*)&sRB[col * LDK + kk + kOff];
                bI[nt] = *(const v2f*)&sIB[col * LDK + kk + kOff];
            }
            #pragma unroll
            for (int mt = 0; mt < 2; ++mt) {
                #pragma unroll
                for (int nt = 0; nt < 2; ++nt) {
                    accR[mt][nt]  = __builtin_amdgcn_wmma_f32_16x16x4_f32(
                        false, aR[mt], false, bR[nt], (short)0, accR[mt][nt],  false, false);
                    accR[mt][nt]  = __builtin_amdgcn_wmma_f32_16x16x4_f32(
                        false, aI[mt], false, bI[nt], (short)0, accR[mt][nt],  false, false);
                    accI1[mt][nt] = __builtin_amdgcn_wmma_f32_16x16x4_f32(
                        false, aI[mt], false, bR[nt], (short)0, accI1[mt][nt], false, false);
                    accI2[mt][nt] = __builtin_amdgcn_wmma_f32_16x16x4_f32(
                        false, aR[mt], false, bI[nt], (short)0, accI2[mt][nt], false, false);
                }
            }
        }
        __syncthreads();
    }

    // ---- epilogue: C/D layout is VGPR j -> M=j (lanes 0-15) / M=j+8 (lanes 16-31), N=lane&15
    const int laneHi8 = (lane >> 4) * 8;
    #pragma unroll
    for (int mt = 0; mt < 2; ++mt) {
        #pragma unroll
        for (int nt = 0; nt < 2; ++nt) {
            int row0 = aBase + waveM * 32 + mt * 16 + laneHi8;
            int col  = cBase + waveN * 32 + nt * 16 + laneLo;
            #pragma unroll
            for (int j = 0; j < 8; ++j) {
                size_t o = ((size_t)b * D_LEN + row0 + j) * D_LEN + col;
                out_r[o] = accR[mt][nt][j];
                out_i[o] = accI1[mt][nt][j] - accI2[mt][nt][j];
            }
        }
    }
}

extern "C" void kernel_launch(void* const* d_in, const int* in_sizes, int n_in,
                              void* d_out, int out_size, void* d_ws, size_t ws_size,
                              hipStream_t stream) {
    const float* in_r = (const float*)d_in[0];
    const float* in_i = (const float*)d_in[1];
    const float* w    = (const float*)d_in[2];
    float* out_r = (float*)d_out;
    float* out_i = out_r + (size_t)32 * D_LEN * D_LEN;   // tuple outputs concatenated

    dim3 grid(D_LEN / 128, D_LEN / 64, 32);  // (n-tiles, m-tiles, batch) = (2,4,32)
    dim3 block(256);
    hipLaunchKernelGGL(complex_mixture_wmma, grid, block, 0, stream,
                       in_r, in_i, w, out_r, out_i);
}